// Recommender_33122787787044
// MI455X (gfx1250) — compile-verified
//
#include <hip/hip_runtime.h>
#include <hip/hip_bf16.h>
#include <math.h>

typedef __attribute__((ext_vector_type(2))) float v2f;
typedef __attribute__((ext_vector_type(8))) float v8f;

#define EMB_D   64
#define MAX_REL 64          // LDS staging capacity (problem has 51 relations)
#define N_HOPS  3

// ---------------------------------------------------------------- utilities
__global__ void k_fill0(float* __restrict__ p, long n) {
    long i = (long)blockIdx.x * blockDim.x + threadIdx.x;
    long s = (long)gridDim.x * blockDim.x;
    for (; i < n; i += s) p[i] = 0.0f;
}

__global__ void k_copy1(const float* __restrict__ src, float* __restrict__ d0, long n) {
    long i = (long)blockIdx.x * blockDim.x + threadIdx.x;
    long s = (long)gridDim.x * blockDim.x;
    for (; i < n; i += s) d0[i] = src[i];
}

__global__ void k_copy2(const float* __restrict__ src, float* __restrict__ d0,
                        float* __restrict__ d1, long n) {
    long i = (long)blockIdx.x * blockDim.x + threadIdx.x;
    long s = (long)gridDim.x * blockDim.x;
    for (; i < n; i += s) { float v = src[i]; d0[i] = v; d1[i] = v; }
}

__global__ void k_count(const int* __restrict__ head, float* __restrict__ cnt, int E) {
    int i = blockIdx.x * blockDim.x + threadIdx.x;
    int s = gridDim.x * blockDim.x;
    for (; i < E; i += s) atomicAdd(&cnt[head[i]], 1.0f);
}

// ------------------------------------------- scatter: msg + segment_sum(agg)
// wave32 per edge; lane owns float2 of the 64-dim row -> 256B coalesced gather.
// Relation table staged in LDS via CDNA5 async global->LDS copy.
__global__ __launch_bounds__(256) void k_scatter(
    const float* __restrict__ ent, const float* __restrict__ rel,
    const int* __restrict__ head, const int* __restrict__ tail,
    const int* __restrict__ etype, float* __restrict__ agg,
    int E, int n_rel)
{
    __shared__ float rel_lds[MAX_REL * EMB_D];
    const int nelem = n_rel * EMB_D;
    for (int i = threadIdx.x; i < nelem; i += blockDim.x) {
#if __has_builtin(__builtin_amdgcn_global_load_async_to_lds_b32) && \
    __has_builtin(__builtin_amdgcn_s_wait_asynccnt)
        __builtin_amdgcn_global_load_async_to_lds_b32(
            (__attribute__((address_space(1))) int*)(rel + i),
            (__attribute__((address_space(3))) int*)(&rel_lds[i]),
            0, 0);
#else
        rel_lds[i] = rel[i];
#endif
    }
#if __has_builtin(__builtin_amdgcn_global_load_async_to_lds_b32) && \
    __has_builtin(__builtin_amdgcn_s_wait_asynccnt)
    __builtin_amdgcn_s_wait_asynccnt(0);
#endif
    __syncthreads();

    const int lane  = threadIdx.x & 31;
    const int wave  = blockIdx.x * (blockDim.x >> 5) + (threadIdx.x >> 5);
    const int nwave = gridDim.x * (blockDim.x >> 5);
    const int c0    = lane * 2;

    for (int e = wave; e < E; e += nwave) {
        const int h = head[e];
        const int t = tail[e];
        const int r = etype[e];
        const float* tp = ent + (size_t)t * EMB_D + c0;
        const float* rp = rel_lds + r * EMB_D + c0;
        float m0 = tp[0] * rp[0];
        float m1 = tp[1] * rp[1];
        float* ap = agg + (size_t)h * EMB_D + c0;
        atomicAdd(ap,     m0);   // agg is L2-resident; atomics resolve at L2
        atomicAdd(ap + 1, m1);
    }
}

// ---------------- entity normalize + residual accumulation (WMMA reduction)
// One wave per 16 rows. Row sum-of-squares via chained V_WMMA_F32_16X16X4_F32
// with an all-ones B matrix: D[m,n] = sum_k A[m,k]. Requires n_rows % 16 == 0
// for the tiles handled here (remainder rows go to k_norm_rows_tail).
__global__ __launch_bounds__(32) void k_norm_ent(
    const float* __restrict__ agg, const float* __restrict__ cnt,
    float* __restrict__ ent_cur, float* __restrict__ ent_res,
    float* __restrict__ drug_res, int n_drugs)
{
    __shared__ float norms[16];
    const int lane = threadIdx.x;               // 0..31, full wave (EXEC all 1s)
    const long tbase = (long)blockIdx.x * 16;
    const int  rowA  = (int)tbase + (lane & 15);    // A-matrix row for this lane
    const int  koff  = (lane >> 4) << 1;            // K sub-offset 0 or 2
    const float invden = 1.0f / fmaxf(cnt[rowA], 1.0f);
    const float* rowp = agg + (size_t)rowA * EMB_D + koff;

#if __has_builtin(__builtin_amdgcn_wmma_f32_16x16x4_f32)
    v8f c = {};
    v2f ones; ones[0] = 1.0f; ones[1] = 1.0f;
    #pragma unroll
    for (int k0 = 0; k0 < EMB_D; k0 += 4) {
        float v0 = rowp[k0]     * invden;
        float v1 = rowp[k0 + 1] * invden;
        v2f a; a[0] = v0 * v0; a[1] = v1 * v1;
        c = __builtin_amdgcn_wmma_f32_16x16x4_f32(
                false, a, false, ones, (short)0, c, false, false);
    }
    // D layout: VGPR r holds row M=r (lanes 0-15) / M=8+r (lanes 16-31); all N equal.
    if (lane == 0) {
        #pragma unroll
        for (int r = 0; r < 8; ++r) norms[r] = c[r];
    } else if (lane == 16) {
        #pragma unroll
        for (int r = 0; r < 8; ++r) norms[8 + r] = c[r];
    }
#else
    float sum = 0.0f;
    #pragma unroll
    for (int k0 = 0; k0 < EMB_D; k0 += 4) {
        float v0 = rowp[k0]     * invden;
        float v1 = rowp[k0 + 1] * invden;
        sum += v0 * v0 + v1 * v1;
    }
    sum += __shfl_xor(sum, 16, 32);     // combine the two K-halves of each row
    if (lane < 16) norms[lane] = sum;
#endif
    __syncthreads();

    // Coalesced write-back: lane owns columns [2*lane, 2*lane+1] of each row.
    #pragma unroll 4
    for (int r = 0; r < 16; ++r) {
        const long row  = tbase + r;
        const float di  = 1.0f / fmaxf(cnt[row], 1.0f);
        const float ni  = 1.0f / fmaxf(sqrtf(norms[r]), 1e-12f);
        const float sc  = di * ni;
        const long idx  = row * EMB_D + lane * 2;
        const float o0  = agg[idx]     * sc;
        const float o1  = agg[idx + 1] * sc;
        ent_cur[idx]     = o0;  ent_cur[idx + 1] = o1;
        ent_res[idx]    += o0;  ent_res[idx + 1] += o1;
        if (row < n_drugs) { drug_res[idx] += o0; drug_res[idx + 1] += o1; }
    }
}

// Remainder rows (n_rows % 16) — wave per row, shuffle reduction (wave32).
__global__ __launch_bounds__(32) void k_norm_rows_tail(
    const float* __restrict__ agg, const float* __restrict__ cnt,
    float* __restrict__ ent_cur, float* __restrict__ ent_res,
    float* __restrict__ drug_res, long row0, int n_drugs)
{
    const long row = row0 + blockIdx.x;
    const int lane = threadIdx.x;
    const long idx = row * EMB_D + lane * 2;
    const float di = 1.0f / fmaxf(cnt[row], 1.0f);
    float v0 = agg[idx] * di, v1 = agg[idx + 1] * di;
    float s = v0 * v0 + v1 * v1;
    for (int off = 16; off; off >>= 1) s += __shfl_xor(s, off, 32);
    const float ni = 1.0f / fmaxf(sqrtf(s), 1e-12f);
    const float o0 = v0 * ni, o1 = v1 * ni;
    ent_cur[idx]  = o0;  ent_cur[idx + 1]  = o1;
    ent_res[idx] += o0;  ent_res[idx + 1] += o1;
    if (row < n_drugs) { drug_res[idx] += o0; drug_res[idx + 1] += o1; }
}

// Relation normalize (in place) + residual; wave per row.
__global__ __launch_bounds__(32) void k_norm_rel(
    float* __restrict__ rel_cur, float* __restrict__ rel_res)
{
    const long row = blockIdx.x;
    const int lane = threadIdx.x;
    const long idx = row * EMB_D + lane * 2;
    float v0 = rel_cur[idx], v1 = rel_cur[idx + 1];
    float s = v0 * v0 + v1 * v1;
    for (int off = 16; off; off >>= 1) s += __shfl_xor(s, off, 32);
    const float ni = 1.0f / fmaxf(sqrtf(s), 1e-12f);
    v0 *= ni; v1 *= ni;
    rel_cur[idx]  = v0;  rel_cur[idx + 1]  = v1;
    rel_res[idx] += v0;  rel_res[idx + 1] += v1;
}

// ---------------------------------------------------------------- launcher
extern "C" void kernel_launch(void* const* d_in, const int* in_sizes, int n_in,
                              void* d_out, int out_size, void* d_ws, size_t ws_size,
                              hipStream_t stream)
{
    const float* entity = (const float*)d_in[0];
    const float* drug   = (const float*)d_in[1];
    const float* rel    = (const float*)d_in[2];
    const int*   eidx   = (const int*)d_in[3];
    const int*   etype  = (const int*)d_in[4];

    const long nEnt  = in_sizes[0] / EMB_D;   // 100000
    const long nDrug = in_sizes[1] / EMB_D;   // 2000
    const long nRel  = in_sizes[2] / EMB_D;   // 51
    const int  E     = in_sizes[3] / 2;       // 1000000
    const int* head  = eidx;
    const int* tail  = eidx + E;

    // workspace carve-out (~51.6 MB): cnt | ent_cur | agg | rel_cur
    float* cnt     = (float*)d_ws;
    float* ent_cur = cnt + nEnt;
    float* agg     = ent_cur + nEnt * EMB_D;
    float* rel_cur = agg + nEnt * EMB_D;

    float* out_ent  = (float*)d_out;
    float* out_drug = out_ent + nEnt * EMB_D;
    float* out_rel  = out_drug + nDrug * EMB_D;

    // ---- init (recomputed every call; deterministic) ----
    k_fill0<<<512, 256, 0, stream>>>(cnt, nEnt);
    k_count<<<2048, 256, 0, stream>>>(head, cnt, E);
    k_copy2<<<4096, 256, 0, stream>>>(entity, ent_cur, out_ent, nEnt * EMB_D);
    k_copy1<<<128, 256, 0, stream>>>(drug, out_drug, nDrug * EMB_D);
    k_copy2<<<8, 256, 0, stream>>>(rel, rel_cur, out_rel, nRel * EMB_D);

    const long tiles = nEnt / 16;
    const long tailRows = nEnt - tiles * 16;

    for (int hop = 0; hop < N_HOPS; ++hop) {
        k_fill0<<<4096, 256, 0, stream>>>(agg, nEnt * EMB_D);
        k_scatter<<<2048, 256, 0, stream>>>(ent_cur, rel_cur, head, tail, etype,
                                            agg, E, (int)nRel);
        if (tiles > 0)
            k_norm_ent<<<(int)tiles, 32, 0, stream>>>(agg, cnt, ent_cur,
                                                      out_ent, out_drug, (int)nDrug);
        if (tailRows > 0)
            k_norm_rows_tail<<<(int)tailRows, 32, 0, stream>>>(agg, cnt, ent_cur,
                                                               out_ent, out_drug,
                                                               tiles * 16, (int)nDrug);
        k_norm_rel<<<(int)nRel, 32, 0, stream>>>(rel_cur, out_rel);
    }
}